// OuterProductMean_52123723104933
// MI455X (gfx1250) — compile-verified
//
#include <hip/hip_runtime.h>

// Shapes (compile-time constants from the reference)
#define R_   256
#define L_   32
#define M_   4
#define INC  64
#define MID  32
#define OUTC 64
#define NN   288   // R_ + L_

typedef float v2f __attribute__((ext_vector_type(2)));
typedef float v8f __attribute__((ext_vector_type(8)));

// ---------------------------------------------------------------------------
// Wave-level 16x16 f32 GEMM tile: D += A[16 x 32] * B^T-ish, where
//   A is row-major [m][k] with leading dim lda (m = lane&15 rows)
//   B supplies column n = lane&15 as a row-major [n][k] array (ldb): this
//   matches V_WMMA_F32_16X16X4_F32's B layout B[k][n] when the "n" data is
//   stored contiguously in k — both operands are contiguous float2 loads.
// K = 32 -> 8 chained v_wmma_f32_16x16x4_f32.
// ---------------------------------------------------------------------------
__device__ __forceinline__ v8f wmma_k32(v8f c,
                                        const float* __restrict__ A, int lda,
                                        const float* __restrict__ B, int ldb) {
  const int t  = threadIdx.x & 31;
  const int mn = t & 15;         // A-row / B-col index for this lane
  const int kh = (t >> 4) << 1;  // K sub-offset (0 or 2)
#pragma unroll
  for (int kk = 0; kk < 32; kk += 4) {
    v2f a = *(const v2f*)(A + mn * lda + kk + kh);
    v2f b = *(const v2f*)(B + mn * ldb + kk + kh);
    c = __builtin_amdgcn_wmma_f32_16x16x4_f32(false, a, false, b,
                                              (short)0, c, false, false);
  }
  return c;
}

// C/D 16x16 f32 layout: VGPR v holds row M = v + 8*(lane>>4), col N = lane&15.
__device__ __forceinline__ void store_tile(float* __restrict__ base, int strideM,
                                           v8f c, float scale,
                                           const float* __restrict__ biasv, int o0) {
  const int t    = threadIdx.x & 31;
  const int n    = t & 15;
  const int half = t >> 4;
  const float bias = biasv[o0 + n];
#pragma unroll
  for (int v = 0; v < 8; ++v) {
    base[(v + half * 8) * strideM + n] = c[v] * scale + bias;
  }
}

// ---------------------------------------------------------------------------
// Stage 1: linear projections  (r_i, r_j: [256][32]; l_i, l_j: [4*32][32])
// ---------------------------------------------------------------------------
__global__ void proj_kernel(const float* __restrict__ rec, const float* __restrict__ lig,
                            const float* __restrict__ r1w, const float* __restrict__ r1b,
                            const float* __restrict__ r2w, const float* __restrict__ r2b,
                            const float* __restrict__ l1w, const float* __restrict__ l1b,
                            const float* __restrict__ l2w, const float* __restrict__ l2b,
                            float* __restrict__ r_i, float* __restrict__ r_j,
                            float* __restrict__ l_i, float* __restrict__ l_j) {
  int id = blockIdx.x * blockDim.x + threadIdx.x;
  if (id < 8192) {                       // r_i
    int i = id >> 5, c = id & 31;
    float acc = r1b[c];
#pragma unroll 8
    for (int x = 0; x < INC; ++x) acc += rec[i * INC + x] * r1w[c * INC + x];
    r_i[id] = acc;
  } else if (id < 16384) {               // r_j
    int t = id - 8192; int i = t >> 5, c = t & 31;
    float acc = r2b[c];
#pragma unroll 8
    for (int x = 0; x < INC; ++x) acc += rec[i * INC + x] * r2w[c * INC + x];
    r_j[t] = acc;
  } else if (id < 20480) {               // l_i  (mi = m*32 + i)
    int t = id - 16384; int mi = t >> 5, c = t & 31;
    float acc = l1b[c];
#pragma unroll 8
    for (int x = 0; x < INC; ++x) acc += lig[mi * INC + x] * l1w[c * INC + x];
    l_i[t] = acc;
  } else if (id < 24576) {               // l_j
    int t = id - 20480; int mi = t >> 5, c = t & 31;
    float acc = l2b[c];
#pragma unroll 8
    for (int x = 0; x < INC; ++x) acc += lig[mi * INC + x] * l2w[c * INC + x];
    l_j[t] = acc;
  }
}

// Stage 1b: mean over the m=4 copies
__global__ void means_kernel(const float* __restrict__ l_i, const float* __restrict__ l_j,
                             float* __restrict__ l_i_m, float* __restrict__ l_j_m) {
  int id = blockIdx.x * blockDim.x + threadIdx.x;
  if (id < 1024) {
    float a = 0.f;
#pragma unroll
    for (int m = 0; m < M_; ++m) a += l_i[m * 1024 + id];
    l_i_m[id] = a * 0.25f;
  } else if (id < 2048) {
    int t = id - 1024;
    float a = 0.f;
#pragma unroll
    for (int m = 0; m < M_; ++m) a += l_j[m * 1024 + t];
    l_j_m[t] = a * 0.25f;
  }
}

// ---------------------------------------------------------------------------
// Stage 2: Out[r][o][y] = sum_x In[r][x] * W[o*1024 + x*32 + y]
// (used for P_rr, P_rl, Q_rl with Rn=256 and S with Rn=128)
// ---------------------------------------------------------------------------
__global__ void pqs_kernel(const float* __restrict__ In, const float* __restrict__ W,
                           float* __restrict__ Out, int Rn) {
  int id = blockIdx.x * blockDim.x + threadIdx.x;
  if (id >= Rn * 2048) return;
  int r = id >> 11;
  int o = (id >> 5) & 63;
  int y = id & 31;
  const float* in = In + r * 32;
  const float* w  = W + o * 1024 + y;
  float acc = 0.f;
#pragma unroll 8
  for (int x = 0; x < 32; ++x) acc += in[x] * w[x * 32];
  Out[id] = acc;   // id == (r*64 + o)*32 + y
}

// ---------------------------------------------------------------------------
// Stage 3: WMMA tile kernels. One wave per 16x16 output tile; 8 waves/block.
// ---------------------------------------------------------------------------
// rr[i,j,o] = sum_y r_j[j,y] * P_rr[i,o,y] + rr_b[o]   -> out[i][j][o]
__global__ void rr_kernel(const float* __restrict__ r_j, const float* __restrict__ P,
                          const float* __restrict__ bias, float* __restrict__ out) {
  int w  = blockIdx.x * (blockDim.x >> 5) + (threadIdx.x >> 5); // 16384 waves
  int i  = w >> 6;
  int j0 = ((w >> 2) & 15) << 4;
  int o0 = (w & 3) << 4;
  v8f c = {0.f, 0.f, 0.f, 0.f, 0.f, 0.f, 0.f, 0.f};
  c = wmma_k32(c, r_j + j0 * 32, 32, P + (i * 64 + o0) * 32, 32);
  store_tile(out + (i * NN + j0) * OUTC + o0, OUTC, c, 1.0f, bias, o0);
}

// rl[i,j,o] -> out[i][256+j][o];  lr[j,i,o] -> out[256+j][i][o]
__global__ void rllr_kernel(const float* __restrict__ l_j_m, const float* __restrict__ l_i_m,
                            const float* __restrict__ P_rl, const float* __restrict__ Q_rl,
                            const float* __restrict__ bias, float* __restrict__ out) {
  int w = blockIdx.x * (blockDim.x >> 5) + (threadIdx.x >> 5);  // 4096 waves
  v8f c = {0.f, 0.f, 0.f, 0.f, 0.f, 0.f, 0.f, 0.f};
  if (w < 2048) {                                   // rl
    int i = w >> 3, j0 = ((w >> 2) & 1) << 4, o0 = (w & 3) << 4;
    c = wmma_k32(c, l_j_m + j0 * 32, 32, P_rl + (i * 64 + o0) * 32, 32);
    store_tile(out + (i * NN + R_ + j0) * OUTC + o0, OUTC, c, 1.0f, bias, o0);
  } else {                                          // lr
    int w2 = w - 2048;
    int ir = w2 >> 3, j0 = ((w2 >> 2) & 1) << 4, o0 = (w2 & 3) << 4;
    c = wmma_k32(c, l_i_m + j0 * 32, 32, Q_rl + (ir * 64 + o0) * 32, 32);
    store_tile(out + ((R_ + j0) * NN + ir) * OUTC + o0, NN * OUTC, c, 1.0f, bias, o0);
  }
}

// ll[i,j,o] = 0.25 * sum_m sum_y l_j[m,j,y] * S[m*32+i,o,y] + ll_b[o]
__global__ void ll_kernel(const float* __restrict__ l_j, const float* __restrict__ S,
                          const float* __restrict__ bias, float* __restrict__ out) {
  int w = blockIdx.x * (blockDim.x >> 5) + (threadIdx.x >> 5);  // 256 waves
  int i = w >> 3, j0 = ((w >> 2) & 1) << 4, o0 = (w & 3) << 4;
  v8f c = {0.f, 0.f, 0.f, 0.f, 0.f, 0.f, 0.f, 0.f};
#pragma unroll
  for (int m = 0; m < M_; ++m) {
    c = wmma_k32(c, l_j + (m * 32 + j0) * 32, 32,
                 S + ((m * 32 + i) * 64 + o0) * 32, 32);
  }
  store_tile(out + ((R_ + i) * NN + R_ + j0) * OUTC + o0, OUTC, c, 0.25f, bias, o0);
}

// ---------------------------------------------------------------------------
extern "C" void kernel_launch(void* const* d_in, const int* in_sizes, int n_in,
                              void* d_out, int out_size, void* d_ws, size_t ws_size,
                              hipStream_t stream) {
  (void)in_sizes; (void)n_in; (void)out_size; (void)ws_size;
  const float* rec = (const float*)d_in[0];
  const float* lig = (const float*)d_in[1];
  /* d_in[2] = pw_rep (zeros, only defines output shape) */
  const float* r1w = (const float*)d_in[3];
  const float* r1b = (const float*)d_in[4];
  const float* r2w = (const float*)d_in[5];
  const float* r2b = (const float*)d_in[6];
  const float* l1w = (const float*)d_in[7];
  const float* l1b = (const float*)d_in[8];
  const float* l2w = (const float*)d_in[9];
  const float* l2b = (const float*)d_in[10];
  const float* rrw = (const float*)d_in[11];
  const float* rrb = (const float*)d_in[12];
  const float* rlw = (const float*)d_in[13];
  const float* rlb = (const float*)d_in[14];
  const float* llw = (const float*)d_in[15];
  const float* llb = (const float*)d_in[16];
  float* out = (float*)d_out;

  // Workspace layout (floats); total 1,861,632 floats (~7.1 MB)
  float* ws    = (float*)d_ws;
  float* r_i   = ws;                 //   8192
  float* r_j   = ws + 8192;          //   8192
  float* l_i   = ws + 16384;         //   4096  [m*32+i][c]
  float* l_j   = ws + 20480;         //   4096
  float* l_i_m = ws + 24576;         //   1024
  float* l_j_m = ws + 25600;         //   1024
  float* P_rr  = ws + 26624;         // 524288  [i][o][y]
  float* P_rl  = P_rr + 524288;      // 524288
  float* Q_rl  = P_rl + 524288;      // 524288
  float* S     = Q_rl + 524288;      // 262144  [m*32+i][o][y]

  proj_kernel<<<96, 256, 0, stream>>>(rec, lig, r1w, r1b, r2w, r2b,
                                      l1w, l1b, l2w, l2b, r_i, r_j, l_i, l_j);
  means_kernel<<<8, 256, 0, stream>>>(l_i, l_j, l_i_m, l_j_m);

  pqs_kernel<<<2048, 256, 0, stream>>>(r_i, rrw, P_rr, 256);
  pqs_kernel<<<2048, 256, 0, stream>>>(r_i, rlw, P_rl, 256);
  pqs_kernel<<<2048, 256, 0, stream>>>(r_j, rlw, Q_rl, 256);
  pqs_kernel<<<1024, 256, 0, stream>>>(l_i, llw, S, 128);

  rr_kernel<<<2048, 256, 0, stream>>>(r_j, P_rr, rrb, out);
  rllr_kernel<<<512, 256, 0, stream>>>(l_j_m, l_i_m, P_rl, Q_rl, rlb, out);
  ll_kernel<<<32, 256, 0, stream>>>(l_j, S, llb, out);
}